// GridSample1d_19851338842351
// MI455X (gfx1250) — compile-verified
//
#include <hip/hip_runtime.h>
#include <hip/hip_bf16.h>
#include <stdint.h>

// GridSample1d: N=64, C=64, L_in=4096, L_out=8192, border padding, align_corners.
// Memory-bound gather (~194 MiB @ 23.3 TB/s ~ 8.5us floor). Strategy:
//  - stage 4 input rows (64KB) per workgroup into LDS via CDNA5 async
//    global->LDS copies (global_load_async_to_lds_b128 / s_wait_asynccnt)
//  - gather both lerp taps from LDS (adjacent reads -> ds_load_2addr_b32)
//  - vectorized b128 grid loads and b128 non-temporal output stores

#define NB    64
#define CB    64
#define LIN   4096
#define LOUT  8192
#define CPB   4          // channels per block -> 4 * 16KB = 64KB LDS
#define TPB   256        // 8 wave32

typedef __attribute__((ext_vector_type(4))) float v4f;

__device__ __forceinline__ void async_load_b128_to_lds(uint32_t lds_byte_addr,
                                                       const void* gptr) {
    uint64_t ga = (uint64_t)(uintptr_t)gptr;
    // GLOBAL_LOAD_ASYNC_TO_LDS_B128: vdst = per-lane LDS byte address,
    // vaddr = per-lane 64-bit global address, saddr = off. Tracked by ASYNCcnt.
    asm volatile("global_load_async_to_lds_b128 %0, %1, off"
                 :: "v"(lds_byte_addr), "v"(ga)
                 : "memory");
}

__device__ __forceinline__ void wait_asynccnt0() {
    asm volatile("s_wait_asynccnt 0" ::: "memory");
}

__global__ __launch_bounds__(TPB) void
GridSample1d_19851338842351_kernel(const float* __restrict__ inp,
                                   const float* __restrict__ grid,
                                   float* __restrict__ out) {
    __shared__ float rows[CPB * LIN];   // 64 KB

    const int tid = threadIdx.x;
    const int n  = blockIdx.x / (CB / CPB);
    const int cg = blockIdx.x % (CB / CPB);
    const int c0 = cg * CPB;

    // ---- Phase 1: async-copy CPB input rows (64 KB) into LDS ----
    const float* src = inp + ((size_t)n * CB + (size_t)c0) * LIN;
    const uint32_t lds_base = (uint32_t)(uintptr_t)(&rows[0]);  // low 32 bits = LDS addr

    const int nvec = (CPB * LIN) / 4;   // 4096 float4 chunks, 16 per thread
    for (int i = tid; i < nvec; i += TPB) {
        async_load_b128_to_lds(lds_base + (uint32_t)i * 16u,
                               (const void*)(src + (size_t)i * 4));
    }
    wait_asynccnt0();     // this wave's async copies landed in LDS
    __syncthreads();      // make all waves' copies visible

    // ---- Phase 2: gather + lerp, 4 consecutive l per thread ----
    const float* gr = grid + (size_t)n * LOUT;
    float* op = out + ((size_t)n * CB + (size_t)c0) * LOUT;

    const float half_span = 0.5f * (float)(LIN - 1);   // align_corners unnormalize

    for (int v = tid; v < LOUT / 4; v += TPB) {
        v4f g = *(const v4f*)(gr + (size_t)v * 4);     // b128 coalesced grid load

        int   i0[4];
        float w0[4], w1[4];
#pragma unroll
        for (int j = 0; j < 4; ++j) {
            float x = (g[j] + 1.0f) * half_span;
            x = fminf(fmaxf(x, 0.0f), (float)(LIN - 1));   // border padding
            // clamp i0 to L-2 so taps are always (i0, i0+1); for x == L-1
            // this yields w1 = 1.0 selecting rows[L-1] -> exact.
            int i = (int)floorf(x);
            i = (i > LIN - 2) ? (LIN - 2) : i;
            i0[j] = i;
            w1[j] = x - (float)i;
            w0[j] = 1.0f - w1[j];
        }

#pragma unroll
        for (int cc = 0; cc < CPB; ++cc) {
            v4f r;
#pragma unroll
            for (int j = 0; j < 4; ++j) {
                // adjacent LDS reads -> ds_load_2addr_b32 fusion
                float v0 = rows[cc * LIN + i0[j]];
                float v1 = rows[cc * LIN + i0[j] + 1];
                r[j] = w0[j] * v0 + w1[j] * v1;
            }
            // streaming write-once output: b128 non-temporal store (512B/wave)
            __builtin_nontemporal_store(r, (v4f*)(op + (size_t)cc * LOUT) + v);
        }
    }
}

extern "C" void kernel_launch(void* const* d_in, const int* in_sizes, int n_in,
                              void* d_out, int out_size, void* d_ws, size_t ws_size,
                              hipStream_t stream) {
    const float* inp  = (const float*)d_in[0];   // [64, 64, 4096] f32
    const float* grid = (const float*)d_in[1];   // [64, 8192]     f32
    float* out = (float*)d_out;                  // [64, 64, 8192] f32

    dim3 blocks(NB * (CB / CPB));                // 1024 workgroups
    GridSample1d_19851338842351_kernel<<<blocks, TPB, 0, stream>>>(inp, grid, out);
}